// BioTokenMucusSim_72782515798418
// MI455X (gfx1250) — compile-verified
//
#include <hip/hip_runtime.h>

// V_WMMA_F32_16X16X4_F32 per-lane register views (wave32):
//   A (16x4)  -> 2 VGPRs: v0 = K0|K2, v1 = K1|K3 (lanes 0-15 | 16-31, M = lane%16)
//   B (4x16)  -> 2 VGPRs: v0 = row K0|K2, v1 = row K1|K3 (N = lane%16)
//   C/D(16x16)-> 8 VGPRs: vr = M=r | M=r+8, N = lane%16
typedef float v2f __attribute__((ext_vector_type(2)));
typedef float v8f __attribute__((ext_vector_type(8)));

#define NB  32   // NUM_BLOCKS
#define WPB 8    // waves (sites) per 256-thread block

__device__ __forceinline__ float clamp01(float x) {
    return __builtin_amdgcn_fmed3f(x, 0.0f, 1.0f);
}

__global__ __launch_bounds__(WPB * 32)
void biotoken_step_kernel(const float* __restrict__ h,
                          const float* __restrict__ W,
                          const float* __restrict__ stim,
                          float* __restrict__ h_out,
                          float* __restrict__ W_out,
                          int num_sites)
{
    __shared__ float4 Bst[WPB][NB][2];    // staged B: row k = [h(k,0..3),1,0,0,0]
    __shared__ float4 hn4[WPB][NB];       // h_new rows
    __shared__ float2 nlb[WPB][NB];       // (|h_new|^2, L_new)
    __shared__ float4 infl[WPB][NB][2];   // [0]=inflow.xyzw  [1].x=rowsum

    const int lane = threadIdx.x & 31;
    const int wv   = threadIdx.x >> 5;
    const int site = blockIdx.x * WPB + wv;
    if (site >= num_sites) return;        // wave-uniform: EXEC stays all-ones

    const int nn = lane & 15;             // M / N index inside a 16-wide tile
    const int hi = lane >> 4;             // which K / M half this lane carries

    const float*  Wsite = W + (size_t)site * (NB * NB);
    const float2* Wp2   = (const float2*)Wsite;

    // ---- per-lane h row (512B contiguous per wave) + stim ----
    float4 hv = ((const float4*)(h + (size_t)site * (NB * 4)))[lane];
    float  st = stim[(size_t)site * NB + lane];

    // stage B operand: lane owns row k=lane, written straight from registers
    Bst[wv][lane][0] = hv;
    Bst[wv][lane][1] = make_float4(1.0f, 0.0f, 0.0f, 0.0f);
    __builtin_amdgcn_wave_barrier();

    // ============ inflow + rowsum: 16x V_WMMA_F32_16X16X4_F32 ==============
    // D(32x16) = W_full(32x32) * B(32x16); self-connection corrected later.
    // Branch-free B reads: col index clamped to 5 (cols 5..7 are zero-filled).
    const float*  bp  = (const float*)&Bst[wv][0][0] + hi * 16 + ((nn < 5) ? nn : 5);
    const float2* pa0 = Wp2 + nn * (NB / 2) + hi;          // rows 0..15
    const float2* pa1 = Wp2 + (nn + 16) * (NB / 2) + hi;   // rows 16..31

    v8f acc0 = {0.f,0.f,0.f,0.f,0.f,0.f,0.f,0.f};
    v8f acc1 = acc0;
    #pragma unroll
    for (int kk = 0; kk < NB; kk += 4) {
        v2f Bv; Bv.x = bp[kk * 8]; Bv.y = bp[kk * 8 + 8];  // const DS offsets
        float2 a0 = pa0[kk >> 1];                          // const IOFFSETs
        float2 a1 = pa1[kk >> 1];
        v2f A0; A0.x = a0.x; A0.y = a0.y;
        v2f A1; A1.x = a1.x; A1.y = a1.y;
        acc0 = __builtin_amdgcn_wmma_f32_16x16x4_f32(false, A0, false, Bv,
                                                     (short)0, acc0, false, false);
        acc1 = __builtin_amdgcn_wmma_f32_16x16x4_f32(false, A1, false, Bv,
                                                     (short)0, acc1, false, false);
    }

    // scatter D columns 0..4 (inflow + rowsum) so lane i owns row i
    if (nn < 5) {
        #pragma unroll
        for (int r = 0; r < 8; ++r) {
            ((float*)&infl[wv][r + 8 * hi][0])[nn]      = acc0[r];
            ((float*)&infl[wv][16 + r + 8 * hi][0])[nn] = acc1[r];
        }
    }
    __builtin_amdgcn_wave_barrier();

    // ============== EPGL pointwise update (self-term corrected) ============
    const float4 iv  = infl[wv][lane][0];
    const float  rs  = infl[wv][lane][1].x;
    const float  wii = Wsite[lane * (NB + 1)];      // diagonal, L0-hot gather
    const float  tw  = (rs - wii) + 1e-8f;
    const float  inv = __builtin_amdgcn_rcpf(tw);
    const float En = (iv.x - wii * hv.x) * inv;
    const float Pn = (iv.y - wii * hv.y) * inv;
    const float Gn = (iv.z - wii * hv.z) * inv;
    const float Ln = (iv.w - wii * hv.w) * inv;

    const float E = hv.x, P = hv.y, G = hv.z, L = hv.w;
    const float E2 = clamp01(E + 0.3f * st - 0.4f * P - 0.2f * G);
    const float P2 = clamp01(P + 0.5f * st + 0.3f * (Pn - P) - 0.2f * E);
    const float G2 = clamp01(G + 0.4f * E * (1.0f - P) + 0.2f * (Gn - G) - 0.3f * P);
    const float gb = 0.5f * En + 0.5f * Gn;
    const float L2 = clamp01(L + 0.4f * gb + 0.3f * (Ln - L) - 0.3f * P);

    float4 hn; hn.x = E2; hn.y = P2; hn.z = G2; hn.w = L2;
    ((float4*)(h_out + (size_t)site * (NB * 4)))[lane] = hn;  // 512B contiguous
    hn4[wv][lane] = hn;
    float2 nl2; nl2.x = E2*E2 + P2*P2 + G2*G2 + L2*L2; nl2.y = L2;
    nlb[wv][lane] = nl2;
    __builtin_amdgcn_wave_barrier();

    // ========= Gram h_new*h_new^T : 4x WMMA (A/B layouts coincide) =========
    const float2* hnp = (const float2*)&hn4[wv][0];
    v2f GA0, GA1;
    {
        float2 t0 = hnp[nn * 2 + hi];            // hn[nn].{xy} | hn[nn].{zw}
        float2 t1 = hnp[(nn + 16) * 2 + hi];
        GA0.x = t0.x; GA0.y = t0.y;
        GA1.x = t1.x; GA1.y = t1.y;
    }
    const v8f zero8 = {0.f,0.f,0.f,0.f,0.f,0.f,0.f,0.f};
    v8f G00 = __builtin_amdgcn_wmma_f32_16x16x4_f32(false, GA0, false, GA0, (short)0, zero8, false, false);
    v8f G01 = __builtin_amdgcn_wmma_f32_16x16x4_f32(false, GA0, false, GA1, (short)0, zero8, false, false);
    v8f G10 = __builtin_amdgcn_wmma_f32_16x16x4_f32(false, GA1, false, GA0, (short)0, zero8, false, false);
    v8f G11 = __builtin_amdgcn_wmma_f32_16x16x4_f32(false, GA1, false, GA1, (short)0, zero8, false, false);

    // ===== W update, r-outer / 4 tiles inner; all const-offset addressing ===
    const float2* nlp = (const float2*)&nlb[wv][0];
    const float2 nlc0 = nlp[nn];
    const float2 nlc1 = nlp[nn + 16];
    const float  sn0  = nlc0.x,          sn1 = nlc1.x;     // column norms
    const float  cl0  = 0.05f * nlc0.y,  cl1 = 0.05f * nlc1.y;

    const float* Wr  = Wsite + (8 * hi) * NB + nn;   // row m0 = 8hi (+r), col nn
    float*       Wor = W_out + (size_t)site * (NB * NB) + (8 * hi) * NB + nn;

    #pragma unroll
    for (int r = 0; r < 8; ++r) {
        const int    m0   = 8 * hi + r;              // row in tiles G00/G01
        const float2 nlm0 = nlp[m0];                 // LDS broadcast per half
        const float2 nlm1 = nlp[m0 + 16];            // row in tiles G10/G11
        const float  cm0  = 0.05f * nlm0.y;
        const float  cm1  = 0.05f * nlm1.y;
        const bool   dg   = (m0 == nn);              // diag hits only (0,0),(1,1)

        float w00 = Wr[r * NB];
        float w01 = Wr[r * NB + 16];
        float w10 = Wr[r * NB + 16 * NB];
        float w11 = Wr[r * NB + 16 * NB + 16];

        float d00 = __builtin_amdgcn_sqrtf(fmaxf(fmaf(-2.0f, G00[r], nlm0.x + sn0), 0.0f));
        float d01 = __builtin_amdgcn_sqrtf(fmaxf(fmaf(-2.0f, G01[r], nlm0.x + sn1), 0.0f));
        float d10 = __builtin_amdgcn_sqrtf(fmaxf(fmaf(-2.0f, G10[r], nlm1.x + sn0), 0.0f));
        float d11 = __builtin_amdgcn_sqrtf(fmaxf(fmaf(-2.0f, G11[r], nlm1.x + sn1), 0.0f));

        float wn00 = clamp01(fmaf(cm0 + cl0, d00, 0.95f * w00));
        float wn01 = clamp01(fmaf(cm0 + cl1, d01, 0.95f * w01));
        float wn10 = clamp01(fmaf(cm1 + cl0, d10, 0.95f * w10));
        float wn11 = clamp01(fmaf(cm1 + cl1, d11, 0.95f * w11));
        if (dg) { wn00 = 0.0f; wn11 = 0.0f; }        // * (1 - eye)

        Wor[r * NB]               = wn00;            // dense 64B row segments
        Wor[r * NB + 16]          = wn01;
        Wor[r * NB + 16 * NB]     = wn10;
        Wor[r * NB + 16 * NB + 16] = wn11;
    }
}

extern "C" void kernel_launch(void* const* d_in, const int* in_sizes, int n_in,
                              void* d_out, int out_size, void* d_ws, size_t ws_size,
                              hipStream_t stream)
{
    const float* h    = (const float*)d_in[0];   // [B,S,32,4]
    const float* W    = (const float*)d_in[1];   // [B,S,32,32]
    const float* stim = (const float*)d_in[2];   // [B,S,32]
    const int num_sites = in_sizes[2] / NB;      // B*S

    float* h_out = (float*)d_out;                            // [B,S,32,4]
    float* W_out = h_out + (size_t)num_sites * NB * 4;       // [B,S,32,32]

    const int blocks = (num_sites + WPB - 1) / WPB;
    biotoken_step_kernel<<<blocks, WPB * 32, 0, stream>>>(
        h, W, stim, h_out, W_out, num_sites);
}